// PartitionedNormalization_16045997818432
// MI455X (gfx1250) — compile-verified
//
#include <hip/hip_runtime.h>

// ---------------------------------------------------------------------------
// Partitioned Normalization for MI455X (gfx1250, wave32).
//
// Stage 1 (pn_stats): segmented sum/sqsum via one-hot WMMA:
//   D[16x16] += A[16x4] x B[4x16], A[m][k] = (domain(row_k)==m),
//   B[k][n] = x[row_k][colTile + n]  (and x^2 for the sqsum accumulator).
//   Counts fall out of the same one-hot values (ccnt += a.x + a.y), folded
//   across lane-halves with one shfl_xor -> no scalar count chains.
//   Loop unrolled 2x (8 rows / iter) so two s_load_b128 of domain ids and
//   four global_load_b32 of x are in flight per iteration.
// Stage 2 (pn_reduce + pn_finalize): deterministic tree reduction of block
//   partials; produce fused scale/bias tables [8][128].
// Stage 3 (pn_apply): streaming pass, 128-bit vectors; x fits in 192MB L2
//   after stage 1, so reads hit L2; output stream + final x reads are
//   non-temporal to avoid churning L2 with one-shot traffic.
// ---------------------------------------------------------------------------

#define BATCH 262144
#define DIM 128
#define NDOM 8
#define EPS 1e-3f

#define S_BLOCKS 256
#define S_THREADS 256                        // 8 waves per block
#define ROWS_PER_BLOCK (BATCH / S_BLOCKS)    // 1024 rows
#define STEPS (ROWS_PER_BLOCK / 4)           // 256 WMMA steps per wave

#define R_BLOCKS 8
#define PB_PER_R (S_BLOCKS / R_BLOCKS)       // 32 partials per reduce block

#define A_BLOCKS 2048
#define A_THREADS 256

typedef __attribute__((ext_vector_type(2))) float v2f;
typedef __attribute__((ext_vector_type(4))) float v4f;
typedef __attribute__((ext_vector_type(8))) float v8f;

// ---------------------------------------------------------------------------
// Stage 1: per-block partial (count, sum, sqsum) via one-hot WMMA.
// ---------------------------------------------------------------------------
__global__ __launch_bounds__(S_THREADS) void pn_stats(
    const float* __restrict__ x, const int* __restrict__ dix,
    float* __restrict__ psum, float* __restrict__ psq, float* __restrict__ pcnt)
{
  const int tid  = threadIdx.x;
  const int wave = tid >> 5;
  const int lane = tid & 31;
  const int half = lane >> 4;     // 0: lanes 0-15 (K=0,1)   1: lanes 16-31 (K=2,3)
  const int ln   = lane & 15;     // A: M index / B: N index
  const int blk  = blockIdx.x;
  const int cb   = wave * 16;     // column tile base
  const int rbase = blk * ROWS_PER_BLOCK;
  const int k0   = half * 2;      // first K slot owned by this lane half
  const int col  = cb + ln;

  v8f acc  = {};                  // D rows 0..7 (lanes 0-15) = per-domain sums
  v8f accq = {};                  // same for sum of squares
  float ccnt = 0.0f;              // per-lane one-hot tally (counts, exact <=2048)

  for (int s = 0; s < STEPS; s += 2) {
    const int r0 = rbase + s * 4;
    // Two 16B uniform loads of domain ids (lowers to s_load_b128 pair).
    const int4 dmA = *reinterpret_cast<const int4*>(dix + r0);
    const int4 dmB = *reinterpret_cast<const int4*>(dix + r0 + 4);

    // Issue all four x loads up front (64B coalesced per lane-half).
    const float bxA = x[(size_t)(r0 + k0)         * DIM + col];
    const float byA = x[(size_t)(r0 + k0 + 1)     * DIM + col];
    const float bxB = x[(size_t)(r0 + 4 + k0)     * DIM + col];
    const float byB = x[(size_t)(r0 + 4 + k0 + 1) * DIM + col];

    // A-matrix 16x4 f32 layout (ISA 7.12.2): lanes 0-15 hold {K0,K1} in
    // VGPR{0,1}, lanes 16-31 hold {K2,K3}.  M == lane index -> one-hot.
    const int d0A = half ? dmA.z : dmA.x;
    const int d1A = half ? dmA.w : dmA.y;
    v2f aA;
    aA.x = (d0A == ln) ? 1.0f : 0.0f;
    aA.y = (d1A == ln) ? 1.0f : 0.0f;

    const int d0B = half ? dmB.z : dmB.x;
    const int d1B = half ? dmB.w : dmB.y;
    v2f aB;
    aB.x = (d0B == ln) ? 1.0f : 0.0f;
    aB.y = (d1B == ln) ? 1.0f : 0.0f;

    ccnt += (aA.x + aA.y) + (aB.x + aB.y);

    // B-matrix 4x16 f32: lanes 0-15 hold rows K0 (VGPR0), K1 (VGPR1) over
    // N=0..15; lanes 16-31 hold K2, K3.
    v2f bA;  bA.x  = bxA;       bA.y  = byA;
    v2f bqA; bqA.x = bxA * bxA; bqA.y = byA * byA;
    v2f bB;  bB.x  = bxB;       bB.y  = byB;
    v2f bqB; bqB.x = bxB * bxB; bqB.y = byB * byB;

    acc  = __builtin_amdgcn_wmma_f32_16x16x4_f32(false, aA, false, bA,
                                                 (short)0, acc,  false, false);
    accq = __builtin_amdgcn_wmma_f32_16x16x4_f32(false, aA, false, bqA,
                                                 (short)0, accq, false, false);
    acc  = __builtin_amdgcn_wmma_f32_16x16x4_f32(false, aB, false, bB,
                                                 (short)0, acc,  false, false);
    accq = __builtin_amdgcn_wmma_f32_16x16x4_f32(false, aB, false, bqB,
                                                 (short)0, accq, false, false);
  }

  // Fold the two lane-halves of the count tally: lane L (<16) gains lane
  // L+16's contribution; lanes 0..7 then hold exact per-domain counts.
  const float ccnt_hi = __shfl_xor(ccnt, 16, 32);
  const float cnt_tot = ccnt + ccnt_hi;

  // D layout: VGPR r, lanes 0-15 -> D[M=r][N=lane]; M=r in 0..7 == domain id.
  // (Lanes 16-31 hold M=8..15 which are identically zero; skip them.)
  if (lane < 16) {
    float* ps = psum + (size_t)blk * NDOM * DIM;
    float* pq = psq  + (size_t)blk * NDOM * DIM;
#pragma unroll
    for (int r = 0; r < NDOM; ++r) {
      ps[r * DIM + cb + lane] = acc[r];
      pq[r * DIM + cb + lane] = accq[r];
    }
  }
  if (wave == 0 && lane < NDOM) {
    pcnt[blk * NDOM + lane] = cnt_tot;
  }
}

// ---------------------------------------------------------------------------
// Stage 2a: 8 blocks each fold 32 block-partials -> intermediate partials.
// ---------------------------------------------------------------------------
__global__ __launch_bounds__(1024) void pn_reduce(
    const float* __restrict__ psum, const float* __restrict__ psq,
    const float* __restrict__ pcnt,
    float* __restrict__ isum, float* __restrict__ isq, float* __restrict__ icnt)
{
  const int t = threadIdx.x;      // 0..1023 == (m, n)
  const int j = blockIdx.x;       // 0..7
  float s = 0.0f, q = 0.0f;
  for (int b = j * PB_PER_R; b < (j + 1) * PB_PER_R; ++b) {
    s += psum[(size_t)b * NDOM * DIM + t];
    q += psq [(size_t)b * NDOM * DIM + t];
  }
  isum[j * NDOM * DIM + t] = s;
  isq [j * NDOM * DIM + t] = q;
  if (t < NDOM) {
    float c = 0.0f;
    for (int b = j * PB_PER_R; b < (j + 1) * PB_PER_R; ++b)
      c += pcnt[b * NDOM + t];
    icnt[j * NDOM + t] = c;
  }
}

// ---------------------------------------------------------------------------
// Stage 2b: final fold + fused scale/bias tables.
//   scale[m][n] = (gg[n]+dg[m][n]) * rsqrt(var + eps)
//   bias [m][n] = (gb[n]+db[m][n]) - mean * scale[m][n]
// ---------------------------------------------------------------------------
__global__ __launch_bounds__(1024) void pn_finalize(
    const float* __restrict__ isum, const float* __restrict__ isq,
    const float* __restrict__ icnt,
    const float* __restrict__ gg, const float* __restrict__ gb,
    const float* __restrict__ dg, const float* __restrict__ db,
    float* __restrict__ scale, float* __restrict__ bias)
{
  const int t = threadIdx.x;      // 0..1023
  const int m = t / DIM;
  const int n = t % DIM;
  float s = 0.0f, q = 0.0f, c = 0.0f;
#pragma unroll
  for (int j = 0; j < R_BLOCKS; ++j) {
    s += isum[j * NDOM * DIM + t];
    q += isq [j * NDOM * DIM + t];
    c += icnt[j * NDOM + m];
  }
  const float nn   = fmaxf(c, 1.0f);
  const float mean = s / nn;
  const float var  = fmaxf(q / nn - mean * mean, 0.0f);
  const float inv  = rsqrtf(var + EPS);
  const float g    = gg[n] + dg[t];
  const float be   = gb[n] + db[t];
  const float sc   = g * inv;
  scale[t] = sc;
  bias[t]  = fmaf(-mean, sc, be);
}

// ---------------------------------------------------------------------------
// Stage 3: streaming apply. One wave per row: 32 lanes x 128-bit == 128 cols.
// Tables (8 KB) staged in LDS; ds_load_b128 per lane.  x loads and out
// stores are non-temporal: both streams are touched exactly once here, so
// keep them from churning the 192MB L2.
// ---------------------------------------------------------------------------
__global__ __launch_bounds__(A_THREADS) void pn_apply(
    const float* __restrict__ x, const int* __restrict__ dix,
    const float* __restrict__ scale, const float* __restrict__ bias,
    float* __restrict__ out)
{
  __shared__ __align__(16) float ls[NDOM * DIM];
  __shared__ __align__(16) float lb[NDOM * DIM];
  for (int i = threadIdx.x; i < NDOM * DIM; i += A_THREADS) {
    ls[i] = scale[i];
    lb[i] = bias[i];
  }
  __syncthreads();

  const int lane = threadIdx.x & 31;
  const int wid  = blockIdx.x * (A_THREADS / 32) + (threadIdx.x >> 5);
  const int nw   = A_BLOCKS * (A_THREADS / 32);

  for (int row = wid; row < BATCH; row += nw) {
    const int d = dix[row];                      // wave-uniform broadcast
    const v4f v = __builtin_nontemporal_load(
        reinterpret_cast<const v4f*>(x + (size_t)row * DIM) + lane);
    const int c = d * DIM + lane * 4;            // 16B-aligned LDS offset
    const v4f s = *reinterpret_cast<const v4f*>(&ls[c]);
    const v4f b = *reinterpret_cast<const v4f*>(&lb[c]);
    v4f o;
    o.x = fmaf(v.x, s.x, b.x);
    o.y = fmaf(v.y, s.y, b.y);
    o.z = fmaf(v.z, s.z, b.z);
    o.w = fmaf(v.w, s.w, b.w);
    __builtin_nontemporal_store(
        o, reinterpret_cast<v4f*>(out + (size_t)row * DIM) + lane);
  }
}

// ---------------------------------------------------------------------------
extern "C" void kernel_launch(void* const* d_in, const int* in_sizes, int n_in,
                              void* d_out, int out_size, void* d_ws, size_t ws_size,
                              hipStream_t stream) {
  (void)in_sizes; (void)n_in; (void)out_size; (void)ws_size;
  const float* x   = (const float*)d_in[0];   // [B, D]
  const float* gg  = (const float*)d_in[1];   // [D]
  const float* gb  = (const float*)d_in[2];   // [D]
  const float* dg  = (const float*)d_in[3];   // [K, D]
  const float* db  = (const float*)d_in[4];   // [K, D]
  const int*   dix = (const int*)d_in[5];     // [B]
  float* out = (float*)d_out;

  // Workspace carving (floats). Total ~2.2 MB; every region is written before
  // it is read (harness poisons ws with 0xAA).
  float* ws    = (float*)d_ws;
  float* psum  = ws;                                   // 256*8*128
  float* psq   = psum + (size_t)S_BLOCKS * NDOM * DIM; // 256*8*128
  float* pcnt  = psq  + (size_t)S_BLOCKS * NDOM * DIM; // 256*8
  float* isum  = pcnt + (size_t)S_BLOCKS * NDOM;       // 8*8*128
  float* isq   = isum + (size_t)R_BLOCKS * NDOM * DIM; // 8*8*128
  float* icnt  = isq  + (size_t)R_BLOCKS * NDOM * DIM; // 8*8
  float* scale = icnt + (size_t)R_BLOCKS * NDOM;       // 8*128
  float* bias  = scale + NDOM * DIM;                   // 8*128

  pn_stats<<<S_BLOCKS, S_THREADS, 0, stream>>>(x, dix, psum, psq, pcnt);
  pn_reduce<<<R_BLOCKS, 1024, 0, stream>>>(psum, psq, pcnt, isum, isq, icnt);
  pn_finalize<<<1, 1024, 0, stream>>>(isum, isq, icnt, gg, gb, dg, db,
                                      scale, bias);
  pn_apply<<<A_BLOCKS, A_THREADS, 0, stream>>>(x, dix, scale, bias, out);
}